// MNGRAPH_90692529422818
// MI455X (gfx1250) — compile-verified
//
#include <hip/hip_runtime.h>
#include <hip/hip_bf16.h>

// Problem constants (match the reference).
#define NN 4096            // N_NODES
#define NE 32768           // N_EDGES
#define SD 4096            // S_DIM
#define CAPE (1 << 19)     // mutable edge-list capacity (grows via reverse-append)
#define CAPN (1 << 19)     // total neighbor-list capacity
#define SMAX (2 * NE + 2)  // max degree score + 1

#define CPB 256            // columns per block in the WMMA mean kernel
#define RPC 4              // neighbor rows per chunk (= WMMA K)

typedef float v2f __attribute__((ext_vector_type(2)));
typedef float v8f __attribute__((ext_vector_type(8)));

#if __has_builtin(__builtin_amdgcn_s_wait_asynccnt)
#define WAIT_ASYNC(n) __builtin_amdgcn_s_wait_asynccnt(n)
#else
#define WAIT_ASYNC(n) asm volatile("s_wait_asynccnt %0" ::"i"(n))
#endif

struct Ws {
  int* meta;       // [0]=K, [1]=m(final edge count), [2]=total neighbors
  int* sel;        // selection order            [NN]
  int* selSorted;  // sorted(select)             [NN]
  int* counts;     // per-hub neighbor count     [NN]
  int* offs;       // CSR offsets into nbr/seg   [NN+1]
  int* fill;       // forward-fill row indices   [NN]
  int* ordv;       // degree-sorted node order   [NN]
  int* visits;     // [NN]
  int* seld;       // selected flag              [NN]
  int* inH;        // scratch membership mask    [NN]
  int* Hlist;      // scratch neighbor list      [NN]
  int* score;      // I+O degree score           [NN]
  int* esrc;       // mutable edge src           [CAPE]
  int* edst;       // mutable edge dst           [CAPE]
  int* nbr;        // flattened hub neighbors    [CAPN]
  int* segA;       // segment id per neighbor    [CAPN]
  int* bucket;     // counting-sort buckets      [SMAX]
};

// ---------------------------------------------------------------------------
// Phase 0: faithful sequential coarsening (single thread), mirroring the
// reference's NumPy while-loop.
// ---------------------------------------------------------------------------
__global__ void coarsen_kernel(const int* __restrict__ ei, Ws w) {
  if (threadIdx.x != 0 || blockIdx.x != 0) return;

  int m = NE;
  for (int e = 0; e < NE; ++e) { w.esrc[e] = ei[e]; w.edst[e] = ei[NE + e]; }
  for (int v = 0; v < NN; ++v) { w.visits[v] = 0; w.seld[v] = 0; w.inH[v] = 0; w.score[v] = 0; }
  for (int e = 0; e < NE; ++e) { w.score[w.esrc[e]]++; w.score[w.edst[e]]++; }

  // Stable counting sort, descending score, ascending index within ties.
  for (int s = 0; s < SMAX; ++s) w.bucket[s] = 0;
  for (int v = 0; v < NN; ++v) w.bucket[w.score[v]]++;
  int start = 0;
  for (int s = SMAX - 1; s >= 0; --s) { int t = w.bucket[s]; w.bucket[s] = start; start += t; }
  for (int v = 0; v < NN; ++v) w.ordv[w.bucket[w.score[v]]++] = v;

  int K = 0;
  int tot = 0;
  w.offs[0] = 0;
  for (int p = 0; p < NN; ++p) {
    int node = w.ordv[p];
    if (w.visits[node]) continue;           // == order[~visits[order]] filtering
    w.sel[K] = node; w.seld[node] = 1; w.visits[node] = 1;

    // H = unique(neighbors of node), sorted ascending (mask scan == np.unique)
    for (int e = 0; e < m; ++e) {
      if (w.esrc[e] == node) w.inH[w.edst[e]] = 1;
      if (w.edst[e] == node) w.inH[w.esrc[e]] = 1;
    }
    int Hc = 0;
    for (int v = 0; v < NN; ++v) if (w.inH[v]) { w.Hlist[Hc++] = v; w.visits[v] = 1; }

    // Relabel Hn (= H \ select) occurrences to `node`
    for (int e = 0; e < m; ++e) {
      int a = w.esrc[e]; if (w.inH[a] && !w.seld[a]) w.esrc[e] = node;
      int b = w.edst[e]; if (w.inH[b] && !w.seld[b]) w.edst[e] = node;
    }
    // Drop self loops
    int wr = 0;
    for (int e = 0; e < m; ++e)
      if (w.esrc[e] != w.edst[e]) { w.esrc[wr] = w.esrc[e]; w.edst[wr] = w.edst[e]; ++wr; }
    m = wr;
    // Append reversed edges whose reverse is absent (checked vs pre-append list)
    int m0 = m;
    for (int e = 0; e < m0; ++e) {
      int a = w.esrc[e], b = w.edst[e];
      bool found = false;
      for (int j = 0; j < m0; ++j)
        if (w.esrc[j] == b && w.edst[j] == a) { found = true; break; }
      if (!found && m < CAPE) { w.esrc[m] = b; w.edst[m] = a; ++m; }
    }
    // Record CSR neighbors
    for (int h = 0; h < Hc; ++h)
      if (tot < CAPN) { w.nbr[tot] = w.Hlist[h]; w.segA[tot] = K; ++tot; }
    w.counts[K] = Hc; ++K; w.offs[K] = tot;
    for (int h = 0; h < Hc; ++h) w.inH[w.Hlist[h]] = 0;
  }

  for (int i = 0; i < K; ++i) w.fill[i] = w.counts[i] ? i : (i ? w.fill[i - 1] : 0);
  int q = 0;
  for (int v = 0; v < NN; ++v) if (w.seld[v]) w.selSorted[q++] = v;
  w.meta[0] = K; w.meta[1] = m; w.meta[2] = tot;
}

// ---------------------------------------------------------------------------
// Phase 1: segment-mean via V_WMMA_F32_16X16X4_F32 with async-to-LDS staging.
//
// One 256-thread block (8 waves) owns a (16-hub x 256-col) tile. Each chunk of
// 4 neighbor rows x 256 cols (4 KB) is staged into LDS with one
// global_load_async_to_lds_b128 per thread (ASYNCcnt, double-buffered:
// issue chunk j+1, s_wait_asynccnt for chunk j, barrier, compute). Each wave
// feeds two 16-col WMMA accumulators from LDS:
//   A(16x4) = one-hot segment selector,  B(4x16) = staged x tile,  C += A*B.
// The loop body is fully unpredicated (clamped indices + cndmask selector),
// so EXEC is all-ones at every WMMA as the ISA requires.
// ---------------------------------------------------------------------------
__global__ __launch_bounds__(256) void mean_wmma_kernel(
    const float* __restrict__ x, const float* __restrict__ W1,
    const float* __restrict__ B1, Ws w, float* __restrict__ out) {
  __shared__ float lds[2][RPC * CPB];                // 2 x 4 KB
  const int K = w.meta[0];
  const int k0 = blockIdx.y * 16;
  if (k0 >= K) return;                               // uniform branch
  const int c0 = blockIdx.x * CPB;
  const int tid  = threadIdx.x;
  const int wave = tid >> 5;
  const int lane = tid & 31;
  const int half = lane >> 4;                        // 0: lanes 0-15, 1: 16-31
  const int mrow = lane & 15;                        // hub row / column index

  const int base = w.offs[k0];
  const int end  = w.offs[min(k0 + 16, K)];
  const int nch  = (end - base + RPC - 1) / RPC;
  const int last = end - 1;                          // valid whenever nch > 0

  // Staging assignment: 4 rows x 64 b128-segments; one async op per thread.
  const int srow  = tid >> 6;                        // 0..3
  const int scol4 = (tid & 63) * 4;                  // 0..252 step 4

  v8f acc0 = {}, acc1 = {};
  const int ct0 = wave * 2;                          // this wave's column tiles

  auto stage = [&](int buf, int ch) {
    const int s   = min(base + ch * RPC + srow, last);   // clamped -> always valid
    const int row = w.nbr[s];
    const float* gptr = x + (size_t)row * SD + c0 + scol4;
    const unsigned ldst = (unsigned)(uintptr_t)&lds[buf][srow * CPB + scol4];
    // GV mode: dsaddr from VDST vgpr, 64-bit global address in VGPR pair.
    asm volatile("global_load_async_to_lds_b128 %0, %1, off"
                 :: "v"(ldst), "v"(gptr) : "memory");
  };

  if (nch > 0) stage(0, 0);
  for (int ch = 0; ch < nch; ++ch) {
    const bool more = (ch + 1 < nch);                // uniform
    if (more) stage((ch + 1) & 1, ch + 1);
    if (more) WAIT_ASYNC(1); else WAIT_ASYNC(0);     // chunk ch staged
    __syncthreads();                                 // visible to all waves

    const float* buf = lds[ch & 1];
    const int j  = base + ch * RPC;
    const int s0 = j + 2 * half;                     // this lane feeds K-slots
    const int s1 = s0 + 1;                           //   2*half and 2*half+1
    const int q0 = min(s0, last), q1 = min(s1, last);
    const int sg0 = w.segA[q0], sg1 = w.segA[q1];    // unpredicated, clamped
    v2f A;
    A.x = (s0 < end && sg0 == k0 + mrow) ? 1.f : 0.f;
    A.y = (s1 < end && sg1 == k0 + mrow) ? 1.f : 0.f;

    v2f Bv0, Bv1;                                    // ds_load from staged tile
    Bv0.x = buf[(2 * half + 0) * CPB + ct0 * 16 + mrow];
    Bv0.y = buf[(2 * half + 1) * CPB + ct0 * 16 + mrow];
    Bv1.x = buf[(2 * half + 0) * CPB + (ct0 + 1) * 16 + mrow];
    Bv1.y = buf[(2 * half + 1) * CPB + (ct0 + 1) * 16 + mrow];

    acc0 = __builtin_amdgcn_wmma_f32_16x16x4_f32(
        false, A, false, Bv0, (short)0, acc0, false, false);
    acc1 = __builtin_amdgcn_wmma_f32_16x16x4_f32(
        false, A, false, Bv1, (short)0, acc1, false, false);

    __syncthreads();                                 // done reading this buffer
  }

  // mean = (W1 * sum + cnt * B1) / max(cnt, 1), into second half of END row.
  for (int t = 0; t < 2; ++t) {
    const v8f acc = t ? acc1 : acc0;
    const int col = c0 + (ct0 + t) * 16 + mrow;
    const float wc = W1[col];
    const float bc = B1[col];
    for (int r = 0; r < 8; ++r) {
      const int hub = k0 + r + 8 * half;             // C/D layout: VGPR r, half
      if (hub >= K) continue;
      const int cnt = w.counts[hub];
      const float denom = (float)max(cnt, 1);
      out[(size_t)hub * (2 * SD) + SD + col] = (wc * acc[r] + (float)cnt * bc) / denom;
    }
  }
}

// Dv half of END: out[k, 0:SD] = x[sel[k]] * W2 + B2   (B128 streams)
__global__ void dv_kernel(const float* __restrict__ x, const float* __restrict__ W2,
                          const float* __restrict__ B2, Ws w, float* __restrict__ out) {
  const int K = w.meta[0];
  const int k = blockIdx.x;
  if (k >= K) return;
  const float4* row = (const float4*)(x + (size_t)w.sel[k] * SD);
  const float4* w2  = (const float4*)W2;
  const float4* b2  = (const float4*)B2;
  float4* dst = (float4*)(out + (size_t)k * (2 * SD));
  for (int c = threadIdx.x; c < SD / 4; c += blockDim.x) {
    float4 r = row[c], a = w2[c], b = b2[c];
    dst[c] = make_float4(r.x * a.x + b.x, r.y * a.y + b.y,
                         r.z * a.z + b.z, r.w * a.w + b.w);
  }
}

// END = END[fill]: rows whose hub had no neighbors copy their forward-fill row.
__global__ void fillcopy_kernel(Ws w, float* __restrict__ out) {
  const int K = w.meta[0];
  const int k = blockIdx.x;
  if (k >= K) return;
  if (w.counts[k] != 0) return;                      // fill[k] == k, nothing to do
  const int s = w.fill[k];
  const float4* src = (const float4*)(out + (size_t)s * (2 * SD));
  float4* dst = (float4*)(out + (size_t)k * (2 * SD));
  for (int c = threadIdx.x; c < (2 * SD) / 4; c += blockDim.x) dst[c] = src[c];
}

__device__ __forceinline__ int lower_bound_i(const int* a, int n, int v) {
  int lo = 0, hi = n;
  while (lo < hi) { int mid = (lo + hi) >> 1; if (a[mid] < v) lo = mid + 1; else hi = mid; }
  return lo;
}

// ei_out = searchsorted(sorted(select), ei), appended after END (values are
// small integers, exact in f32; d_out is exposed as float*).
__global__ void edges_kernel(Ws w, float* __restrict__ out) {
  const int K = w.meta[0];
  const int m = w.meta[1];
  const int e = blockIdx.x * blockDim.x + threadIdx.x;
  if (e >= m) return;
  float* oe = out + (size_t)K * (2 * SD);
  oe[e]     = (float)lower_bound_i(w.selSorted, K, w.esrc[e]);
  oe[m + e] = (float)lower_bound_i(w.selSorted, K, w.edst[e]);
}

// ---------------------------------------------------------------------------
extern "C" void kernel_launch(void* const* d_in, const int* in_sizes, int n_in,
                              void* d_out, int out_size, void* d_ws, size_t ws_size,
                              hipStream_t stream) {
  const float* x  = (const float*)d_in[0];
  const int*   ei = (const int*)  d_in[1];   // edge_index (2, NE)
  const float* W1 = (const float*)d_in[2];
  const float* W2 = (const float*)d_in[3];
  const float* B1 = (const float*)d_in[4];
  const float* B2 = (const float*)d_in[5];
  float* out = (float*)d_out;

  char* p = (char*)d_ws;
  auto carve = [&](size_t elems) -> int* {
    int* r = (int*)p;
    p += (elems * sizeof(int) + 255) & ~(size_t)255;
    return r;
  };
  Ws w;
  w.meta = carve(16);       w.sel = carve(NN);      w.selSorted = carve(NN);
  w.counts = carve(NN);     w.offs = carve(NN + 1); w.fill = carve(NN);
  w.ordv = carve(NN);       w.visits = carve(NN);   w.seld = carve(NN);
  w.inH = carve(NN);        w.Hlist = carve(NN);    w.score = carve(NN);
  w.esrc = carve(CAPE);     w.edst = carve(CAPE);
  w.nbr = carve(CAPN);      w.segA = carve(CAPN);   w.bucket = carve(SMAX);
  // total ws use: ~9 MB

  hipLaunchKernelGGL(coarsen_kernel, dim3(1), dim3(1), 0, stream, ei, w);
  hipLaunchKernelGGL(mean_wmma_kernel, dim3(SD / CPB, NN / 16), dim3(256), 0, stream,
                     x, W1, B1, w, out);
  hipLaunchKernelGGL(dv_kernel, dim3(NN), dim3(256), 0, stream, x, W2, B2, w, out);
  hipLaunchKernelGGL(fillcopy_kernel, dim3(NN), dim3(256), 0, stream, w, out);
  hipLaunchKernelGGL(edges_kernel, dim3((CAPE + 255) / 256), dim3(256), 0, stream, w, out);
}